// Transform_52158082843453
// MI455X (gfx1250) — compile-verified
//
#include <hip/hip_runtime.h>
#include <math.h>

// ---------------------------------------------------------------------------
// Problem constants (match reference)
// ---------------------------------------------------------------------------
#define NN      50000
#define EE      800000
#define F_IN    256
#define CC      64
#define NEG_SLOPE 0.01f
#define INV_SQRT_C 0.125f   // 1/sqrt(64)

typedef __attribute__((ext_vector_type(2))) float v2f;
typedef __attribute__((ext_vector_type(8))) float v8f;

__device__ __forceinline__ float leaky(float v) {
    return v > 0.0f ? v : NEG_SLOPE * v;
}

// Order-preserving float atomic max via integer atomics (m init'd to -inf).
__device__ __forceinline__ void atomicMaxFloat(float* addr, float val) {
    if (val >= 0.0f) {
        atomicMax((int*)addr, __float_as_int(val));
    } else {
        atomicMin((unsigned int*)addr, __float_as_uint(val));
    }
}

// ---------------------------------------------------------------------------
// Kernel 1: h0 = leaky_relu(x[N,256] @ Wf[256,64] + bf)
// One wave per 16x16 output tile; 4 waves/block cover the 64 output columns.
// K = 256 -> 64 chained v_wmma_f32_16x16x4_f32 per tile. N % 16 == 0, so no
// divergence: EXEC stays all-ones (WMMA requirement).
// ---------------------------------------------------------------------------
__global__ void __launch_bounds__(128)
gemm_in_kernel(const float* __restrict__ x, const float* __restrict__ Wf,
               const float* __restrict__ bf, float* __restrict__ h) {
    const int lane = threadIdx.x & 31;
    const int wave = threadIdx.x >> 5;          // 0..3 -> column tile
    const int row0 = blockIdx.x * 16;           // 3125 blocks * 16 = 50000
    const int col0 = wave * 16;
    const int l15  = lane & 15;
    const int khalf = (lane >> 4) << 1;         // 0 or 2 (K-pair per lane half)

    v8f acc = {};
    const float* xrow = x + (size_t)(row0 + l15) * F_IN;
    const int col = col0 + l15;

#pragma unroll 4
    for (int k0 = 0; k0 < F_IN; k0 += 4) {
        const int kk = k0 + khalf;
        // A fragment: 16x4 f32, lane L<16 holds (M=L, K0..K1), L>=16 (M=L-16, K2..K3)
        v2f a = *(const v2f*)(xrow + kk);
        // B fragment: 4x16 f32, mirrored layout (N=lane&15, same K pairing)
        v2f b;
        b.x = Wf[(size_t)kk * CC + col];
        b.y = Wf[(size_t)(kk + 1) * CC + col];
        acc = __builtin_amdgcn_wmma_f32_16x16x4_f32(false, a, false, b,
                                                    (short)0, acc, false, false);
    }

    // C/D layout: VGPR r -> row r (lanes 0-15) / row r+8 (lanes 16-31)
    const int rbase = row0 + ((lane >> 4) << 3);
    const float bias = bf[col];
#pragma unroll
    for (int r = 0; r < 8; ++r) {
        h[(size_t)(rbase + r) * CC + col] = leaky(acc[r] + bias);
    }
}

// ---------------------------------------------------------------------------
// Kernel 2: qkvs[N,256] = h[N,64] @ Wcat[64,256] + bcat
// Wcat = [Wq | Wk | Wv | Wskip] packed once. 16 waves/block cover 256 cols.
// K = 64 -> 16 chained WMMAs per tile.
// ---------------------------------------------------------------------------
__global__ void __launch_bounds__(512)
gemm_qkvs_kernel(const float* __restrict__ h, const float* __restrict__ Wcat,
                 const float* __restrict__ bcat, float* __restrict__ qkvs) {
    const int lane = threadIdx.x & 31;
    const int wave = threadIdx.x >> 5;          // 0..15 -> column tile
    const int row0 = blockIdx.x * 16;
    const int col0 = wave * 16;
    const int l15  = lane & 15;
    const int khalf = (lane >> 4) << 1;

    v8f acc = {};
    const float* hrow = h + (size_t)(row0 + l15) * CC;
    const int col = col0 + l15;

#pragma unroll
    for (int k0 = 0; k0 < CC; k0 += 4) {
        const int kk = k0 + khalf;
        v2f a = *(const v2f*)(hrow + kk);
        v2f b;
        b.x = Wcat[(size_t)kk * 256 + col];
        b.y = Wcat[(size_t)(kk + 1) * 256 + col];
        acc = __builtin_amdgcn_wmma_f32_16x16x4_f32(false, a, false, b,
                                                    (short)0, acc, false, false);
    }

    const int rbase = row0 + ((lane >> 4) << 3);
    const float bias = bcat[col];
#pragma unroll
    for (int r = 0; r < 8; ++r) {
        qkvs[(size_t)(rbase + r) * 256 + col] = acc[r] + bias;
    }
}

// ---------------------------------------------------------------------------
// Pack [Wq|Wk|Wv|Wskip] -> Wcat[64][256], biases -> bcat[256]
// ---------------------------------------------------------------------------
__global__ void pack_weights_kernel(const float* __restrict__ Wq, const float* __restrict__ bq,
                                    const float* __restrict__ Wk, const float* __restrict__ bk,
                                    const float* __restrict__ Wv, const float* __restrict__ bv,
                                    const float* __restrict__ Wskip, const float* __restrict__ bskip,
                                    float* __restrict__ Wcat, float* __restrict__ bcat) {
    const int tid = blockIdx.x * blockDim.x + threadIdx.x;   // 64*256 threads
    if (tid >= CC * 256) return;
    const int k = tid >> 8;        // input dim 0..63
    const int j = tid & 255;       // output col 0..255
    float w, b;
    if (j < 64)       { w = Wq[k * CC + j];            b = bq[j]; }
    else if (j < 128) { w = Wk[k * CC + (j - 64)];     b = bk[j - 64]; }
    else if (j < 192) { w = Wv[k * CC + (j - 128)];    b = bv[j - 128]; }
    else              { w = Wskip[k * CC + (j - 192)]; b = bskip[j - 192]; }
    Wcat[tid] = w;
    if (k == 0) bcat[j] = b;
}

// ---------------------------------------------------------------------------
// Per-layer init: m = -inf, den = 0, num = 0
// ---------------------------------------------------------------------------
__global__ void init_layer_kernel(float* __restrict__ num, float* __restrict__ den,
                                  float* __restrict__ m) {
    const int tid = blockIdx.x * blockDim.x + threadIdx.x;
    if (tid < NN * CC) {
        num[tid] = 0.0f;
    } else if (tid < NN * CC + NN) {
        den[tid - NN * CC] = 0.0f;
    } else if (tid < NN * CC + 2 * NN) {
        m[tid - NN * CC - NN] = -INFINITY;
    }
}

// ---------------------------------------------------------------------------
// Edge pass 1 (wave per edge): alpha = dot(q[dst], k[src] + e) / sqrt(C)
// e recomputed on the fly: e_c = ea0*We[0][c] + ea1*We[1][c]
// Each 64-float q/k slice of a qkvs row is 256 B -> a wave's 64-channel read is fully coalesced.
// ---------------------------------------------------------------------------
__global__ void __launch_bounds__(256)
edge_alpha_kernel(const int* __restrict__ ei, const float* __restrict__ edge_attr,
                  const float* __restrict__ We, const float* __restrict__ qkvs,
                  float* __restrict__ alpha, float* __restrict__ m) {
    const int lane = threadIdx.x & 31;
    const int e = blockIdx.x * 8 + (threadIdx.x >> 5);
    if (e >= EE) return;
    const int s = ei[e];
    const int d = ei[EE + e];
    const float ea0 = edge_attr[(size_t)e * 2 + 0];
    const float ea1 = edge_attr[(size_t)e * 2 + 1];

    const float* qr = qkvs + (size_t)d * 256;         // q at cols [0,64)
    const float* kr = qkvs + (size_t)s * 256 + 64;    // k at cols [64,128)

    float part = 0.0f;
#pragma unroll
    for (int c = lane; c < CC; c += 32) {
        const float ec = fmaf(ea0, We[c], ea1 * We[CC + c]);
        part = fmaf(qr[c], kr[c] + ec, part);
    }
#pragma unroll
    for (int off = 16; off > 0; off >>= 1) part += __shfl_xor(part, off, 32);

    if (lane == 0) {
        const float a = part * INV_SQRT_C;
        alpha[e] = a;
        atomicMaxFloat(&m[d], a);
    }
}

// ---------------------------------------------------------------------------
// Edge pass 2 (wave per edge): ex = exp(alpha - m[dst]);
// den[dst] += ex; num[dst][c] += ex * (v[src][c] + e_c)
// Softmax normalization deferred to the node kernel (saves one edge pass).
// ---------------------------------------------------------------------------
__global__ void __launch_bounds__(256)
edge_scatter_kernel(const int* __restrict__ ei, const float* __restrict__ edge_attr,
                    const float* __restrict__ We, const float* __restrict__ qkvs,
                    const float* __restrict__ alpha, const float* __restrict__ m,
                    float* __restrict__ num, float* __restrict__ den) {
    const int lane = threadIdx.x & 31;
    const int e = blockIdx.x * 8 + (threadIdx.x >> 5);
    if (e >= EE) return;
    const int s = ei[e];
    const int d = ei[EE + e];
    const float ea0 = edge_attr[(size_t)e * 2 + 0];
    const float ea1 = edge_attr[(size_t)e * 2 + 1];

    const float ex = __expf(alpha[e] - m[d]);
    if (lane == 0) atomicAdd(&den[d], ex);

    const float* vr = qkvs + (size_t)s * 256 + 128;   // v at cols [128,192)
    float* nr = num + (size_t)d * CC;
#pragma unroll
    for (int c = lane; c < CC; c += 32) {
        const float ec = fmaf(ea0, We[c], ea1 * We[CC + c]);
        atomicAdd(&nr[c], ex * (vr[c] + ec));
    }
}

// ---------------------------------------------------------------------------
// Node update: h_new = leaky_relu(num/den + skip)   (skip at qkvs cols [192,256))
// ---------------------------------------------------------------------------
__global__ void __launch_bounds__(256)
node_update_kernel(const float* __restrict__ num, const float* __restrict__ den,
                   const float* __restrict__ qkvs, float* __restrict__ hout) {
    const int tid = blockIdx.x * blockDim.x + threadIdx.x;
    if (tid >= NN * CC) return;
    const int n = tid >> 6;
    const int c = tid & 63;
    const float dn = den[n];
    const float agg = dn > 0.0f ? num[tid] / dn : 0.0f;
    hout[tid] = leaky(agg + qkvs[(size_t)n * 256 + 192 + c]);
}

// ---------------------------------------------------------------------------
// Host launcher
// ---------------------------------------------------------------------------
extern "C" void kernel_launch(void* const* d_in, const int* in_sizes, int n_in,
                              void* d_out, int out_size, void* d_ws, size_t ws_size,
                              hipStream_t stream) {
    const float* x         = (const float*)d_in[0];
    const int*   ei        = (const int*)  d_in[1];
    const float* edge_attr = (const float*)d_in[2];
    const float* Wf        = (const float*)d_in[3];
    const float* bf        = (const float*)d_in[4];
    const float* Wq        = (const float*)d_in[5];
    const float* bq        = (const float*)d_in[6];
    const float* Wk        = (const float*)d_in[7];
    const float* bk        = (const float*)d_in[8];
    const float* Wv        = (const float*)d_in[9];
    const float* bv        = (const float*)d_in[10];
    const float* We        = (const float*)d_in[11];
    const float* Wskip     = (const float*)d_in[12];
    const float* bskip     = (const float*)d_in[13];

    // Workspace carving (floats). Everything that is read is fully written
    // first within this call -> deterministic despite 0xAA poisoning.
    float* ws    = (float*)d_ws;
    float* h     = ws;                      size_t off = (size_t)NN * CC;
    float* qkvs  = ws + off;                off += (size_t)NN * 256;
    float* num   = ws + off;                off += (size_t)NN * CC;
    float* m     = ws + off;                off += NN;
    float* den   = ws + off;                off += NN;
    float* alpha = ws + off;                off += EE;
    float* Wcat  = ws + off;                off += (size_t)CC * 256;
    float* bcat  = ws + off;                off += 256;

    // One-time: pack fused weight matrix and compute h0.
    pack_weights_kernel<<<(CC * 256 + 255) / 256, 256, 0, stream>>>(
        Wq, bq, Wk, bk, Wv, bv, Wskip, bskip, Wcat, bcat);
    gemm_in_kernel<<<NN / 16, 128, 0, stream>>>(x, Wf, bf, h);

    const int edgeBlocks = (EE + 7) / 8;              // 8 edges (waves) / block
    const int initBlocks = (NN * CC + 2 * NN + 255) / 256;
    const int nodeBlocks = (NN * CC + 255) / 256;

    const float* hin = h;
    for (int layer = 0; layer < 3; ++layer) {
        gemm_qkvs_kernel<<<NN / 16, 512, 0, stream>>>(hin, Wcat, bcat, qkvs);
        init_layer_kernel<<<initBlocks, 256, 0, stream>>>(num, den, m);
        edge_alpha_kernel<<<edgeBlocks, 256, 0, stream>>>(ei, edge_attr, We, qkvs,
                                                          alpha, m);
        edge_scatter_kernel<<<edgeBlocks, 256, 0, stream>>>(ei, edge_attr, We, qkvs,
                                                            alpha, m, num, den);
        float* hout = (layer == 2) ? (float*)d_out : h;
        node_update_kernel<<<nodeBlocks, 256, 0, stream>>>(num, den, qkvs, hout);
        hin = hout;
    }
}